// AGG_BP_APPR_49168785605031
// MI455X (gfx1250) — compile-verified
//
#include <hip/hip_runtime.h>
#include <hip/hip_bf16.h>

// APPR propagation: out = x + 0.15 * segment_sum(x[src], dst)
//   x: [N=100000, D=64] f32, edge_index: [2, E=1600000] int32
// Memory-bound: x (25.6MB) and out (25.6MB) both fit in MI455X's 192MB L2, so
// the random gather and atomic scatter run at L2 speed; HBM floor ~80MB.
// CDNA5 paths: async global->LDS tile staging (ASYNCcnt) double-buffered,
// readfirstlane scalarization for saddr-form gathers, hw f32 atomics in L2.

#define APPR_WEIGHT 0.15f
#define D_FEAT 64
#define EDGES_PER_BLOCK 256
#define TILES_PER_BLOCK 8

// ---- CDNA5 async global->LDS copy (ASYNCcnt-tracked), GVS addressing ------
__device__ __forceinline__ void async_load_b32_to_lds(unsigned lds_addr,
                                                      unsigned gbyte_off,
                                                      const void* sbase) {
    asm volatile("global_load_async_to_lds_b32 %0, %1, %2"
                 :: "v"(lds_addr), "v"(gbyte_off), "s"(sbase)
                 : "memory");
}

__device__ __forceinline__ void wait_asynccnt0() {
#if defined(__has_builtin) && __has_builtin(__builtin_amdgcn_s_wait_asynccnt)
    __builtin_amdgcn_s_wait_asynccnt(0);
#else
    asm volatile("s_wait_asynccnt 0" ::: "memory");
#endif
}

// Hardware f32 atomic add, no-return form (STOREcnt only, resolves in L2).
// NOTE: deliberately NO "memory" clobber — `out` is never read in this kernel,
// so the compiler may freely schedule x-gathers / LDS reads across pending
// atomics (asm volatile still keeps the atomics themselves in order & alive).
__device__ __forceinline__ void atomic_add_f32(float* p, float v) {
    asm volatile("global_atomic_add_f32 %0, %1, off"
                 :: "v"(p), "v"(v));
}

// ---------------------------- kernel 1: out = x ----------------------------
__global__ __launch_bounds__(256) void appr_init_kernel(
    const float* __restrict__ x, float* __restrict__ out, int n4) {
    int i = blockIdx.x * blockDim.x + threadIdx.x;
    if (i < n4) {
        reinterpret_cast<float4*>(out)[i] =
            reinterpret_cast<const float4*>(x)[i];
    }
}

__global__ void appr_tail_kernel(const float* __restrict__ x,
                                 float* __restrict__ out, int start, int nd) {
    int i = start + threadIdx.x;
    if (i < nd) out[i] = x[i];
}

// ------------------- kernel 2: atomic scatter of messages ------------------
// One wave per edge: lane l handles features [2l, 2l+1] (float2 => one
// coalesced 256B request per wave). Each block owns TILES_PER_BLOCK tiles of
// 256 edges; (src,dst) index tiles are double-buffered in LDS via async copy.
__global__ __launch_bounds__(256) void appr_edge_kernel(
    const float* __restrict__ x, const int* __restrict__ edge_index,
    float* __restrict__ out, int n_edges, int n_tiles) {
    __shared__ int lds_idx[2][2][EDGES_PER_BLOCK];  // [buf][src|dst][edge]

    const int t     = threadIdx.x;
    const int tile0 = blockIdx.x * TILES_PER_BLOCK;
    if (tile0 >= n_tiles) return;
    int tile_end = tile0 + TILES_PER_BLOCK;
    if (tile_end > n_tiles) tile_end = n_tiles;

    const int wave = t >> 5;
    const int lane = t & 31;
    const int f    = lane << 1;   // feature pair offset
    const int ebeg = wave << 5;   // this wave's 32 edges within a tile

    // Stage one tile's (src,dst) indices into LDS buffer `buf` (clamped tail;
    // clamped entries are staged but never processed).
    auto stage = [&](int buf, int tile) {
        int e  = tile * EDGES_PER_BLOCK + t;
        int ec = (e < n_edges) ? e : (n_edges - 1);
        unsigned off_src = (unsigned)ec * 4u;
        unsigned off_dst = ((unsigned)n_edges + (unsigned)ec) * 4u;
        // Generic pointer low 32 bits == workgroup-relative LDS byte address.
        unsigned la = (unsigned)(size_t)(void*)&lds_idx[buf][0][t];
        unsigned lb = (unsigned)(size_t)(void*)&lds_idx[buf][1][t];
        async_load_b32_to_lds(la, off_src, edge_index);
        async_load_b32_to_lds(lb, off_dst, edge_index);
    };

    auto process_edge = [&](int buf, int el) {
        // Uniform LDS read (broadcast) -> scalarize for saddr-form gather.
        int s = __builtin_amdgcn_readfirstlane(lds_idx[buf][0][el]);
        int d = __builtin_amdgcn_readfirstlane(lds_idx[buf][1][el]);
        float2 v = *reinterpret_cast<const float2*>(
            x + (size_t)s * D_FEAT + f);
        float* o = out + (size_t)d * D_FEAT + f;
        atomic_add_f32(o,     APPR_WEIGHT * v.x);
        atomic_add_f32(o + 1, APPR_WEIGHT * v.y);
    };

    stage(0, tile0);          // prologue: prefetch first tile
    int buf = 0;

    for (int tile = tile0; tile < tile_end; ++tile) {
        wait_asynccnt0();     // this wave's pending stage (tile) landed
        __syncthreads();      // all waves staged `buf`; prior tile fully read

        // Prefetch next tile into the other buffer while we process this one.
        if (tile + 1 < tile_end) stage(buf ^ 1, tile + 1);

        int base  = tile * EDGES_PER_BLOCK;
        int n_loc = n_edges - base - ebeg;
        if (n_loc > 32) n_loc = 32;

        if (n_loc == 32) {
#pragma unroll 8
            for (int k = 0; k < 32; ++k) process_edge(buf, ebeg + k);
        } else {
            for (int k = 0; k < n_loc; ++k) process_edge(buf, ebeg + k);
        }
        buf ^= 1;
    }
}

// ------------------------------- launcher ----------------------------------
extern "C" void kernel_launch(void* const* d_in, const int* in_sizes, int n_in,
                              void* d_out, int out_size, void* d_ws, size_t ws_size,
                              hipStream_t stream) {
    const float* x   = (const float*)d_in[0];
    const int*   ei  = (const int*)d_in[1];
    float*       out = (float*)d_out;

    const int nd      = in_sizes[0];       // N * D
    const int n_edges = in_sizes[1] / 2;   // [2, E]
    const int n4      = nd >> 2;
    const int rem     = nd & 3;

    int init_blocks = (n4 + 255) / 256;
    appr_init_kernel<<<init_blocks, 256, 0, stream>>>(x, out, n4);
    if (rem) {
        appr_tail_kernel<<<1, 4, 0, stream>>>(x, out, n4 * 4, nd);
    }

    int n_tiles = (n_edges + EDGES_PER_BLOCK - 1) / EDGES_PER_BLOCK;
    int edge_blocks = (n_tiles + TILES_PER_BLOCK - 1) / TILES_PER_BLOCK;
    appr_edge_kernel<<<edge_blocks, 256, 0, stream>>>(x, ei, out, n_edges,
                                                      n_tiles);
}